// BERT_CRF_60782377173502
// MI455X (gfx1250) — compile-verified
//
#include <hip/hip_runtime.h>

typedef float v2f __attribute__((ext_vector_type(2)));
typedef float v4f __attribute__((ext_vector_type(4)));
typedef float v8f __attribute__((ext_vector_type(8)));

#define NB      64      // batch
#define NS      512     // seq len
#define NH      768     // hidden
#define NT      12      // tags
#define NM      (NB * NS)   // 32768 rows
#define NEGV    (-10000.0f)
#define START_IDX 9
#define STOP_IDX  10
#define WPITCH  772     // 768 + 4 floats pad -> LDS bank stride 4, conflict-free

// ---------------------------------------------------------------------------
// Kernel 1: feats[M,12] = E[M,768] @ W[12,768]^T + b   via V_WMMA_F32_16X16X4_F32
// 256 threads = 8 waves; each wave owns one 16-row output tile (N padded to 16).
// K processed in blocks of 8: one b128 global load (A) + one b128 LDS load (B)
// feeds two WMMAs.
// A frag (16x4 f32): lanes 0-15 = rows, VGPR0/1 = K+0/K+1 ; lanes 16-31 = K+2/K+3
// B frag (4x16 f32): lanes 0-15 = cols, VGPR0/1 = K+0/K+1 ; lanes 16-31 = K+2/K+3
// C/D (16x16 f32): VGPR r -> M = r + 8*(lane>=16), N = lane&15
// ---------------------------------------------------------------------------
__global__ __launch_bounds__(256) void feats_wmma_kernel(
    const float* __restrict__ E, const float* __restrict__ W,
    const float* __restrict__ bias, float* __restrict__ feats)
{
    __shared__ float Wlds[16 * WPITCH];
    const int tid = threadIdx.x;

    // zero LDS (rows 12..15 must be exactly 0), then stage W
    for (int i = tid; i < 16 * WPITCH; i += 256) Wlds[i] = 0.0f;
    __syncthreads();
    for (int i = tid; i < NT * NH; i += 256) {
        int t = i / NH, h = i - t * NH;
        Wlds[t * WPITCH + h] = W[i];
    }
    __syncthreads();

    const int wave = tid >> 5;
    const int lane = tid & 31;
    const int hi   = lane >> 4;        // 0: lanes 0-15, 1: lanes 16-31
    const int l16  = lane & 15;
    const int rowBase = blockIdx.x * 128 + wave * 16;

    const float* pA = E + (size_t)(rowBase + l16) * NH + hi * 4;
    const float* pB = &Wlds[l16 * WPITCH + hi * 4];

    v8f c = {};
    for (int t = 0; t < NH / 8; ++t) {
        v4f a4 = *(const v4f*)(pA + t * 8);   // E[row][8t + 4*hi .. +3]
        v4f b4 = *(const v4f*)(pB + t * 8);   // Wlds[n][8t + 4*hi .. +3]
        v2f a01 = {a4.x, a4.y};
        v2f b01 = {b4.x, b4.y};
        c = __builtin_amdgcn_wmma_f32_16x16x4_f32(false, a01, false, b01,
                                                  (short)0, c, false, false);
        v2f a23 = {a4.z, a4.w};
        v2f b23 = {b4.z, b4.w};
        c = __builtin_amdgcn_wmma_f32_16x16x4_f32(false, a23, false, b23,
                                                  (short)0, c, false, false);
    }

    const int n = l16;
    if (n < NT) {
        const float bn = bias[n];
        #pragma unroll
        for (int r = 0; r < 8; ++r) {
            int m = rowBase + r + 8 * hi;
            feats[(size_t)m * NT + n] = c[r] + bn;
        }
    }
}

// ---------------------------------------------------------------------------
// Kernel 2: Viterbi forward + backtrace. One wave32 per batch element.
// lane j holds fv[j]; broadcast via __shfl; transition row cached in regs;
// backpointers in LDS (512*12 bytes); lane 0 backtraces from LDS.
// ---------------------------------------------------------------------------
__global__ __launch_bounds__(32) void viterbi_kernel(
    const float* __restrict__ feats, const float* __restrict__ trans,
    float* __restrict__ out)
{
    __shared__ unsigned char bp[NS * NT];
    const int b    = blockIdx.x;
    const int lane = threadIdx.x;

    float tr[NT];
    float tstop;
    if (lane < NT) {
        #pragma unroll
        for (int j = 0; j < NT; ++j) tr[j] = trans[lane * NT + j];
        tstop = trans[STOP_IDX * NT + lane];
    } else {
        #pragma unroll
        for (int j = 0; j < NT; ++j) tr[j] = NEGV;
        tstop = NEGV;
    }

    float fv = (lane == START_IDX) ? 0.0f : NEGV;
    const float* fb = feats + (size_t)b * NS * NT;

    for (int s = 0; s < NS; ++s) {
        float feat = (lane < NT) ? fb[s * NT + lane] : 0.0f;
        float best = -3.0e38f;
        int bestj = 0;
        #pragma unroll
        for (int j = 0; j < NT; ++j) {
            float v = __shfl(fv, j, 32) + tr[j];
            if (v > best) { best = v; bestj = j; }   // strict '>' == argmax-first
        }
        if (lane < NT) bp[s * NT + lane] = (unsigned char)bestj;
        fv = best + feat;
    }
    __syncthreads();

    float term = (lane < NT) ? (fv + tstop) : -3.0e38f;
    float best = -3.0e38f;
    int bestj = 0;
    #pragma unroll
    for (int j = 0; j < NT; ++j) {
        float tj = __shfl(term, j, 32);
        if (tj > best) { best = tj; bestj = j; }
    }

    if (lane == 0) {
        out[b] = best;                       // path_score [B,1]
        float* seq = out + NB + (size_t)b * NS;
        int cur = bestj;
        for (int s = NS - 1; s >= 0; --s) {  // reverse scan: emit carry, then hop
            seq[s] = (float)cur;
            cur = bp[s * NT + cur];
        }
    }
}

extern "C" void kernel_launch(void* const* d_in, const int* in_sizes, int n_in,
                              void* d_out, int out_size, void* d_ws, size_t ws_size,
                              hipStream_t stream) {
    const float* E     = (const float*)d_in[0];   // [64,512,768]
    const float* W     = (const float*)d_in[1];   // [12,768]
    const float* bias  = (const float*)d_in[2];   // [12]
    const float* trans = (const float*)d_in[3];   // [12,12] (pre-masked in setup)
    float* out   = (float*)d_out;                 // [64] scores + [64*512] tags
    float* feats = (float*)d_ws;                  // 32768*12 floats = 1.5 MB

    feats_wmma_kernel<<<NM / 128, 256, 0, stream>>>(E, W, bias, feats);
    viterbi_kernel<<<NB, 32, 0, stream>>>(feats, trans, out);
}